// MatrixNetwork_46505905881152
// MI455X (gfx1250) — compile-verified
//
#include <hip/hip_runtime.h>
#include <math.h>

// ---------------------------------------------------------------------------
// MI455X (gfx1250) implementation of the 4096-step matrix-chain recurrence.
//
//   v <- token_mats[tid] @ v   (4096 sequential steps, right-to-left tokens)
//   v <- base_mat @ v ; v <- v/||v|| ; out = decode_vecs @ v
//
// Only 13 distinct 1536^2 fp32 matrices (123 MB) -> fully L2-resident
// (192 MB L2). The chain is sync-latency bound: ONE persistent cooperative
// kernel, 96 WGs x 512 threads; WG owns a 16-row tile of v; 16 waves split
// K=1536 into 96-wide slices; each 16x4 GEMV tile is a
// V_WMMA_F32_16X16X4_F32 (exact fp32) with v broadcast across all 16 B cols.
//
// Pipelining: the step-s A-tiles depend only on token_ids (known at once),
// while v_s is ready only after the step barrier. Each step issues ALL its
// A loads into registers first, THEN spins on the previous step's slot; the
// barrier wait hides the L2 load latency. After release: b loads + WMMAs on
// TWO independent accumulator chains (halves the serial D->C accumulation
// depth), LDS cross-wave reduce, publish tile, arrive.
// ---------------------------------------------------------------------------

typedef float v2f __attribute__((ext_vector_type(2)));
typedef float v8f __attribute__((ext_vector_type(8)));

#define NDIM   1536
#define SLEN   4096
#define VOCAB  13
#define NWG    96      // one 16-row tile per workgroup: 96*16 = 1536
#define TPB    512     // 16 waves (wave32)
#define WAVES  16
#define KSLICE 96      // K columns per wave: 16*96 = 1536
#define KITER  24      // KSLICE / 4

// scratch layout
//   float    vbuf[2][NDIM]      ping-pong vector buffers
//   unsigned slots[SLEN+1]      per-step barrier arrival counters
__global__ __launch_bounds__(256) void chain_init_kernel(
    const float* __restrict__ query, float* __restrict__ vbuf,
    unsigned* __restrict__ slots) {
  int i = blockIdx.x * blockDim.x + threadIdx.x;
  if (i < SLEN + 1) slots[i] = 0u;
  if (i < NDIM) vbuf[i] = query[i];  // buffer 0 seeded with q
}

__device__ __forceinline__ void spin_on(unsigned* slot) {
  while (__hip_atomic_load(slot, __ATOMIC_ACQUIRE,
                           __HIP_MEMORY_SCOPE_AGENT) < (unsigned)NWG) {
    __builtin_amdgcn_s_sleep(1);
  }
}

__global__ __launch_bounds__(TPB) void chain_kernel(
    const int* __restrict__ token_ids, const float* __restrict__ base_mat,
    const float* __restrict__ token_mats, const float* __restrict__ decode_vecs,
    float* __restrict__ vbuf, unsigned* __restrict__ slots,
    float* __restrict__ out) {
  const int wg   = blockIdx.x;          // row tile: rows [wg*16, wg*16+16)
  const int tid  = threadIdx.x;
  const int w    = tid >> 5;            // wave id 0..15 -> K slice
  const int lane = tid & 31;
  const int m    = lane & 15;           // row within tile (A layout: lane=M)
  const int hi   = lane >> 4;           // half-wave: K {0,1} vs {2,3}
  const int row  = wg * 16 + m;
  const int koff = w * KSLICE + 2 * hi; // this lane's K base offset

  __shared__ float red[WAVES * 16];     // per-wave partial row sums
  __shared__ float wsum[WAVES];
  __shared__ float sinv;

  for (int s = 0; s <= SLEN; ++s) {
    // step s < SLEN applies token matrix (flipped order); step SLEN = base_mat
    const float* Amat = (s < SLEN)
        ? (token_mats + (size_t)token_ids[SLEN - 1 - s] * NDIM * NDIM)
        : base_mat;
    const float* vc   = vbuf + (size_t)(s & 1) * NDIM;
    float*       vn   = vbuf + (size_t)((s + 1) & 1) * NDIM;
    const float* arow = Amat + (size_t)row * NDIM + koff;

    // ---- issue ALL A-tile loads for this step (independent of barrier) ----
    // A 16x4 fp32 layout: lanes 0-15 -> K = kb,kb+1 ; lanes 16-31 -> kb+2,kb+3
    v2f a[KITER];
#pragma unroll
    for (int j = 0; j < KITER; ++j) a[j] = *(const v2f*)(arow + 4 * j);

    // ---- wait for v_s (previous step's writers); overlaps A-load latency ----
    if (s > 0) {
      if (tid == 0) spin_on(&slots[s - 1]);
      __syncthreads();
    }

    // ---- b = v broadcast across all 16 B columns (mirrored K split) ----
    const float* bp = vc + koff;
    v2f b[KITER];
#pragma unroll
    for (int j = 0; j < KITER; ++j) b[j] = *(const v2f*)(bp + 4 * j);

    // two independent accumulator chains (even/odd k-tiles)
    v8f acc0 = {0.f, 0.f, 0.f, 0.f, 0.f, 0.f, 0.f, 0.f};
    v8f acc1 = {0.f, 0.f, 0.f, 0.f, 0.f, 0.f, 0.f, 0.f};
#pragma unroll
    for (int j = 0; j < KITER; j += 2) {
      acc0 = __builtin_amdgcn_wmma_f32_16x16x4_f32(
          false, a[j], false, b[j], (short)0, acc0, false, false);
      acc1 = __builtin_amdgcn_wmma_f32_16x16x4_f32(
          false, a[j + 1], false, b[j + 1], (short)0, acc1, false, false);
    }
    v8f acc = acc0 + acc1;

    // D layout: VGPR r -> M=r (lanes 0-15) / M=r+8 (lanes 16-31); all N
    // columns are identical (broadcast B), so lane 0 / lane 16 suffice.
    if (lane == 0) {
#pragma unroll
      for (int r = 0; r < 8; ++r) red[w * 16 + r] = acc[r];
    } else if (lane == 16) {
#pragma unroll
      for (int r = 0; r < 8; ++r) red[w * 16 + 8 + r] = acc[r];
    }
    __syncthreads();
    if (tid < 16) {  // reduce K slices across the 16 waves, publish tile
      float sum = 0.f;
#pragma unroll
      for (int ww = 0; ww < WAVES; ++ww) sum += red[ww * 16 + tid];
      vn[wg * 16 + tid] = sum;
    }
    __syncthreads();

    // ---- arrive at this step's slot (one arrival per WG) ----
    if (tid == 0) {
      __threadfence();  // make vn tile visible at device scope
      __hip_atomic_fetch_add(&slots[s], 1u, __ATOMIC_RELEASE,
                             __HIP_MEMORY_SCOPE_AGENT);
    }
    __syncthreads();
  }

  if (wg != 0) return;

  // ---- epilogue (WG0 only): wait for final step, normalize + decode ----
  if (tid == 0) spin_on(&slots[SLEN]);
  __syncthreads();

  const float* v = vbuf + (size_t)((SLEN + 1) & 1) * NDIM;  // buffer 1
  float ss = 0.f;
  for (int i = tid; i < NDIM; i += TPB) ss += v[i] * v[i];
#pragma unroll
  for (int off = 16; off; off >>= 1) ss += __shfl_xor(ss, off, 32);
  if (lane == 0) wsum[w] = ss;
  __syncthreads();
  if (tid == 0) {
    float t = 0.f;
#pragma unroll
    for (int i = 0; i < WAVES; ++i) t += wsum[i];
    sinv = 1.0f / (sqrtf(t) + 1e-12f);
  }
  __syncthreads();
  if (w < VOCAB) {  // one wave per vocab row
    const float* drow = decode_vecs + (size_t)w * NDIM;
    float d = 0.f;
    for (int i = lane; i < NDIM; i += 32) d += drow[i] * v[i];
#pragma unroll
    for (int off = 16; off; off >>= 1) d += __shfl_xor(d, off, 32);
    if (lane == 0) out[w] = d * sinv;
  }
}

extern "C" void kernel_launch(void* const* d_in, const int* in_sizes, int n_in,
                              void* d_out, int out_size, void* d_ws,
                              size_t ws_size, hipStream_t stream) {
  const int*   token_ids   = (const int*)d_in[0];
  const float* base_mat    = (const float*)d_in[1];
  const float* token_mats  = (const float*)d_in[2];
  const float* decode_vecs = (const float*)d_in[3];
  const float* query       = (const float*)d_in[4];
  float*       out         = (float*)d_out;

  float*    vbuf  = (float*)d_ws;                                   // 2*NDIM
  unsigned* slots = (unsigned*)((char*)d_ws + 2 * NDIM * sizeof(float));

  (void)in_sizes; (void)n_in; (void)out_size; (void)ws_size;

  // stream-ordered init each launch (d_ws is poisoned / stale between runs)
  int init_n = (SLEN + 1 > NDIM) ? (SLEN + 1) : NDIM;
  chain_init_kernel<<<(init_n + 255) / 256, 256, 0, stream>>>(query, vbuf,
                                                              slots);
  chain_kernel<<<NWG, TPB, 0, stream>>>(token_ids, base_mat, token_mats,
                                        decode_vecs, vbuf, slots, out);
}